// DiscreteContinuousConv2d_87170656239867
// MI455X (gfx1250) — compile-verified
//
#include <hip/hip_runtime.h>

typedef float v2f __attribute__((ext_vector_type(2)));
typedef float v8f __attribute__((ext_vector_type(8)));

#define N_PTS   16384      // N_IN == N_OUT
#define KSZ     9
#define BC      64         // B*C = 2*32
#define RED     288        // K*C reduction depth

// ---------------- Phase 0: zero the xk accumulator (9*16384*64 floats) ------
__global__ void zero_xk_kernel(float4* __restrict__ xk4) {
    int tid = blockIdx.x * blockDim.x + threadIdx.x;   // 2,359,296 threads
    xk4[tid] = make_float4(0.f, 0.f, 0.f, 0.f);
}

// ---------------- Phase 1: xq[n*64 + f] = x[f*N + n] * qw[n],  f = b*32+c ---
__global__ void xq_kernel(const float* __restrict__ x,
                          const float* __restrict__ qw,
                          float* __restrict__ xq) {
    int tid = blockIdx.x * blockDim.x + threadIdx.x;   // 1,048,576 threads
    int n = tid >> 6;
    int f = tid & 63;
    xq[tid] = x[(size_t)f * N_PTS + n] * qw[n];
}

// ------------- Phase 1b: repack weight into WMMA B-operand pair layout ------
// r = k*32 + c  (reduction index, 0..287);  wp2[((r>>1)*32 + o)*2 + (r&1)]
__global__ void repack_w_kernel(const float* __restrict__ w,   // (O,C,K)
                                float* __restrict__ wp2) {
    int tid = blockIdx.x * blockDim.x + threadIdx.x;   // 9216 threads
    int o = tid & 31;
    int r = tid >> 5;                                   // 0..287
    int k = r >> 5;
    int c = r & 31;
    wp2[(((r >> 1) * 32 + o) << 1) + (r & 1)] = w[((o << 5) + c) * KSZ + k];
}

// ---------------- Phase 2: sparse gather + atomic scatter-add ---------------
// 16 threads per nnz entry; each handles 4 contiguous floats of the 64-row.
__global__ void scatter_kernel(const float* __restrict__ xq,
                               const float* __restrict__ psi,
                               const int* __restrict__ idx_k,
                               const int* __restrict__ idx_out,
                               const int* __restrict__ idx_in,
                               float* __restrict__ xk, int nnz) {
    int tid = blockIdx.x * blockDim.x + threadIdx.x;
    int e = tid >> 4;
    if (e >= nnz) return;
    int j = (tid & 15) << 2;
    float p = psi[e];
    const float4 v = *(const float4*)(xq + (size_t)idx_in[e] * BC + j);
    float* dst = xk + ((size_t)idx_k[e] * N_PTS + idx_out[e]) * BC + j;
    unsafeAtomicAdd(dst + 0, p * v.x);
    unsafeAtomicAdd(dst + 1, p * v.y);
    unsafeAtomicAdd(dst + 2, p * v.z);
    unsafeAtomicAdd(dst + 3, p * v.w);
}

// ---------------- Phase 3: WMMA f32 16x16x4 GEMM + bias ---------------------
// One wave per 16(n) x 16(o) tile per batch b; reduction depth 288 (72 WMMAs).
__global__ void gemm_wmma_kernel(const float* __restrict__ xk,
                                 const float* __restrict__ wp2,
                                 const float* __restrict__ bias,
                                 float* __restrict__ out) {
    int lane = threadIdx.x & 31;
    int w = blockIdx.x * (blockDim.x >> 5) + (threadIdx.x >> 5);  // 0..4095
    int b     = w >> 11;            // batch
    int rem   = w & 2047;
    int oBase = (rem >> 10) << 4;   // 0 or 16
    int nBase = (rem & 1023) << 4;  // grid-point tile base
    int m   = lane & 15;
    int hi  = lane >> 4;            // 0 or 1
    int cOff = hi << 1;             // A/B K-pair offset per ISA VGPR layout

    v8f acc = {};
    // B pointer walks the repacked weights linearly: +128 floats / WMMA step
    const float* bptr = wp2 + (size_t)((hi * 32 + oBase + m) << 1);

    for (int k = 0; k < KSZ; ++k) {
        const float* aptr = xk + ((size_t)(k * N_PTS + nBase + m)) * BC
                               + (b << 5) + cOff;
        #pragma unroll
        for (int c0 = 0; c0 < 32; c0 += 4) {
            v2f a  = *(const v2f*)(aptr + c0);
            v2f bm = *(const v2f*)(bptr);
            acc = __builtin_amdgcn_wmma_f32_16x16x4_f32(
                false, a, false, bm, (short)0, acc, false, false);
            bptr += 128;
        }
    }

    int o = oBase + m;
    float bv = bias[o];
    float* op = out + ((size_t)((b << 5) + o)) * N_PTS + nBase + (hi << 3);
    #pragma unroll
    for (int v = 0; v < 8; ++v) op[v] = acc[v] + bv;
}

extern "C" void kernel_launch(void* const* d_in, const int* in_sizes, int n_in,
                              void* d_out, int out_size, void* d_ws, size_t ws_size,
                              hipStream_t stream) {
    const float* x    = (const float*)d_in[0];
    const float* qw   = (const float*)d_in[1];
    const float* psi  = (const float*)d_in[2];
    const float* wgt  = (const float*)d_in[3];
    const float* bias = (const float*)d_in[4];
    const int*   ik   = (const int*)d_in[5];
    const int*   io   = (const int*)d_in[6];
    const int*   ii   = (const int*)d_in[7];
    int nnz = in_sizes[2];
    float* out = (float*)d_out;

    char* ws = (char*)d_ws;
    float* xq  = (float*)(ws);                        // 4 MB
    float* wp2 = (float*)(ws + (4u << 20));           // 36 KB
    float* xk  = (float*)(ws + (8u << 20));           // 36 MB

    zero_xk_kernel<<<(KSZ * N_PTS * BC / 4) / 256, 256, 0, stream>>>((float4*)xk);
    xq_kernel<<<(N_PTS * BC) / 256, 256, 0, stream>>>(x, qw, xq);
    repack_w_kernel<<<(RED * 32) / 256, 256, 0, stream>>>(wgt, wp2);
    scatter_kernel<<<(nnz * 16 + 255) / 256, 256, 0, stream>>>(
        xq, psi, ik, io, ii, xk, nnz);
    gemm_wmma_kernel<<<512, 256, 0, stream>>>(xk, wp2, bias, out);
}